// Attention_17016660426876
// MI455X (gfx1250) — compile-verified
//
#include <hip/hip_runtime.h>
#include <math.h>

typedef __attribute__((ext_vector_type(2))) float v2f;
typedef __attribute__((ext_vector_type(8))) float v8f;
typedef unsigned int u32x4 __attribute__((ext_vector_type(4)));
typedef int i32x4 __attribute__((ext_vector_type(4)));
typedef int i32x8 __attribute__((ext_vector_type(8)));

#define BB 4
#define NN 2048
#define CC 1024
#define HH 16
#define DD 64
#define BN (BB*NN)               // 8192 rows
#define FQKV (3*CC)              // 3072
#define BHN (BB*HH*NN)           // 131072 head-rows
#define QKV_ELEMS ((size_t)BB*HH*NN*DD)   // 8388608 floats per tensor

#if defined(__gfx1250__) && __has_builtin(__builtin_amdgcn_tensor_load_to_lds)
#define USE_TDM 1
#else
#define USE_TDM 0
#endif

__device__ __forceinline__ v8f wmma4(v2f a, v2f b, v8f c) {
  // V_WMMA_F32_16X16X4_F32: full fp32 matrix op, D = A(16x4) x B(4x16) + C
  return __builtin_amdgcn_wmma_f32_16x16x4_f32(false, a, false, b, (short)0, c,
                                               false, false);
}

__device__ __forceinline__ v8f v8f_zero() {
  v8f z = {0.f,0.f,0.f,0.f,0.f,0.f,0.f,0.f};
  return z;
}

#if USE_TDM
__device__ __forceinline__ unsigned lds_off(const void* p) {
  // generic pointer to a __shared__ object: low 32 bits = LDS byte address
  return (unsigned)(unsigned long long)(uintptr_t)p;
}

// 2-D tile DMA: rows x width fp32 elements, global row stride `stride` elems,
// LDS destination padded by (pad_amt+1) dwords every 2^(pad_int+1) dwords.
__device__ __forceinline__ void tdm_load_2d(unsigned ldsaddr, const float* gptr,
                                            unsigned width, unsigned rows,
                                            unsigned stride, unsigned pad_int,
                                            unsigned pad_amt) {
  unsigned long long ga = (unsigned long long)(uintptr_t)gptr;
  u32x4 g0;
  g0.x = 1u;                                  // count=1, user descriptor
  g0.y = ldsaddr;                             // lds_addr
  g0.z = (unsigned)ga;                        // global_addr[31:0]
  g0.w = (unsigned)(ga >> 32) | 0x80000000u;  // global_addr[56:32] | type=2
  i32x8 g1;
  g1[0] = (int)((2u << 16)            // data_size = 4 bytes
              | (1u << 20)            // pad_enable
              | (pad_int << 22)       // pad_interval code
              | (pad_amt << 25));     // pad_amount code
  g1[1] = (int)(width << 16);         // tensor_dim0[15:0]
  g1[2] = (int)(rows << 16);          // tensor_dim0 hi=0 | tensor_dim1[15:0]
  g1[3] = (int)(width << 16);         // tensor_dim1 hi=0 | tile_dim0
  g1[4] = (int)rows;                  // tile_dim1 | tile_dim2=0
  g1[5] = (int)stride;                // tensor_dim0_stride[31:0]
  g1[6] = 0;                          // stride hi | tensor_dim1_stride lo
  g1[7] = 0;
  i32x4 z4 = {0, 0, 0, 0};
#if __clang_major__ >= 23
  i32x8 z8 = {0, 0, 0, 0, 0, 0, 0, 0};
  __builtin_amdgcn_tensor_load_to_lds(g0, g1, z4, z4, z8, 0);
#else
  __builtin_amdgcn_tensor_load_to_lds(g0, g1, z4, z4, 0);
#endif
}
#endif  // USE_TDM

// ---------------------------------------------------------------------------
// Tiled fp32 GEMM:  D[m,n] = sum_k A[m,k] * W[n,k]   (A: BNxK row-major,
// W: FxK row-major, K = CC = 1024).
// MODE 0: scatter store into q/k/v workspaces laid out (B,H,N,D).
// MODE 1: plain row-major store (B*N, CC).
// Workgroup tile 64(M) x 128(N); K staged 32-deep, double-buffered in LDS via
// TDM (wave 0 issues next slab, s_wait_tensorcnt pipelines it). 8 waves 4x2.
// ---------------------------------------------------------------------------
template<int MODE>
__global__ __launch_bounds__(256)
void gemm_kernel(const float* __restrict__ A, const float* __restrict__ W,
                 float* __restrict__ out0, float* __restrict__ out1,
                 float* __restrict__ out2) {
  __shared__ __align__(16) float As[2][64 * 36];   // stride 36 = TDM pad
  __shared__ __align__(16) float Bs[2][128 * 36];

  const int tid  = threadIdx.x;
  const int wave = tid >> 5;
  const int lane = tid & 31;
  const int l16  = lane & 15;
  const int half = lane >> 4;
  const int wm   = wave & 3;       // 4 M-slices of 16
  const int wn   = wave >> 2;      // 2 N-slices of 64
  const int m0   = blockIdx.x * 64;
  const int n0   = blockIdx.y * 128;

  v8f acc[4];
#pragma unroll
  for (int t = 0; t < 4; ++t) acc[t] = v8f_zero();

#if USE_TDM
  if (wave == 0) {   // prime buffer 0 (TDM ignores EXEC; one issue per wave)
    tdm_load_2d(lds_off(&As[0][0]), &A[(size_t)m0 * CC], 32, 64, CC, 4, 3);
    tdm_load_2d(lds_off(&Bs[0][0]), &W[(size_t)n0 * CC], 32, 128, CC, 4, 3);
  }
#endif

  for (int k0 = 0; k0 < CC; k0 += 32) {
    const int cur = (k0 >> 5) & 1;
#if USE_TDM
    if (wave == 0) {
      if (k0 + 32 < CC) {   // async-prefetch next slab into the other buffer
        tdm_load_2d(lds_off(&As[cur ^ 1][0]),
                    &A[(size_t)m0 * CC + k0 + 32], 32, 64, CC, 4, 3);
        tdm_load_2d(lds_off(&Bs[cur ^ 1][0]),
                    &W[(size_t)n0 * CC + k0 + 32], 32, 128, CC, 4, 3);
        __builtin_amdgcn_s_wait_tensorcnt(2);  // in-order: current slab done
      } else {
        __builtin_amdgcn_s_wait_tensorcnt(0);
      }
    }
#else
#pragma unroll
    for (int i = 0; i < 2; ++i) {
      int idx = tid + i * 256;
      int r = idx >> 3, c4 = (idx & 7) << 2;
      *reinterpret_cast<float4*>(&As[cur][r * 36 + c4]) =
          *reinterpret_cast<const float4*>(&A[(size_t)(m0 + r) * CC + k0 + c4]);
    }
#pragma unroll
    for (int i = 0; i < 4; ++i) {
      int idx = tid + i * 256;
      int r = idx >> 3, c4 = (idx & 7) << 2;
      *reinterpret_cast<float4*>(&Bs[cur][r * 36 + c4]) =
          *reinterpret_cast<const float4*>(&W[(size_t)(n0 + r) * CC + k0 + c4]);
    }
    if (k0 + 32 < CC) {
      __builtin_prefetch(&A[(size_t)(m0 + (tid >> 2)) * CC + k0 + 32], 0, 1);
      __builtin_prefetch(&W[(size_t)(n0 + (tid >> 1)) * CC + k0 + 32], 0, 1);
    }
#endif
    __syncthreads();

#pragma unroll
    for (int kk = 0; kk < 32; kk += 4) {
      const int kh = kk + 2 * half;
      // A frag: lane -> M = l16, K = kh, kh+1
      v2f a = *reinterpret_cast<const v2f*>(&As[cur][(wm * 16 + l16) * 36 + kh]);
#pragma unroll
      for (int t = 0; t < 4; ++t) {
        // B frag: lane -> N = l16, K = kh, kh+1  (B(k,n) = W[n,k])
        v2f b = *reinterpret_cast<const v2f*>(
            &Bs[cur][(wn * 64 + t * 16 + l16) * 36 + kh]);
        acc[t] = wmma4(a, b, acc[t]);
      }
    }
    __syncthreads();   // all reads done before buffer `cur` is refilled
  }

  // Epilogue. C layout: VGPR r -> row r (lanes 0-15) / row r+8 (lanes 16-31).
#pragma unroll
  for (int t = 0; t < 4; ++t) {
#pragma unroll
    for (int r = 0; r < 8; ++r) {
      const int m = m0 + wm * 16 + r + half * 8;
      const int n = n0 + wn * 64 + t * 16 + l16;
      const float val = acc[t][r];
      if (MODE == 0) {
        // f = s*1024 + h*64 + d ;  scatter into (B,H,N,D)
        const int s  = n >> 10;
        const int hh = (n >> 6) & 15;
        const int d  = n & 63;
        const int b  = m >> 11;
        const int ns = m & 2047;
        float* dst = (s == 0) ? out0 : (s == 1) ? out1 : out2;
        dst[((((size_t)b * HH + hh) * NN) + ns) * DD + d] = val;
      } else {
        out0[(size_t)m * CC + n] = val;
      }
    }
  }
}

// ---------------------------------------------------------------------------
// Fused RMSNorm (fp32, eps=1e-6) + interleaved RoPE on q and k, in place.
// One wave per (b,h,n) row of 64; 2 elements/lane; wave32 shuffle reduction.
// ---------------------------------------------------------------------------
__global__ __launch_bounds__(256)
void rmsrope_kernel(float* __restrict__ q, float* __restrict__ k,
                    const float* __restrict__ cosb,
                    const float* __restrict__ sinb,
                    const float* __restrict__ qg, const float* __restrict__ kg) {
  const int wave = threadIdx.x >> 5;
  const int lane = threadIdx.x & 31;
  const long long row = (long long)blockIdx.x * 8 + wave;  // 0 .. 2*BHN-1
  const bool isK = row >= (long long)BHN;
  const long long rt = isK ? row - (long long)BHN : row;
  float* base = (isK ? k : q) + rt * DD;
  const float* g = isK ? kg : qg;
  const int d0 = lane * 2;

  v2f x = *reinterpret_cast<const v2f*>(base + d0);
  float ss = x.x * x.x + x.y * x.y;
  ss += __shfl_xor(ss, 1);  ss += __shfl_xor(ss, 2);
  ss += __shfl_xor(ss, 4);  ss += __shfl_xor(ss, 8);
  ss += __shfl_xor(ss, 16);
  const float norm = rsqrtf(ss * (1.0f / 64.0f) + 1e-6f);

  const int n = (int)(rt % NN);
  v2f c = *reinterpret_cast<const v2f*>(cosb + (size_t)n * DD + d0);
  v2f s = *reinterpret_cast<const v2f*>(sinb + (size_t)n * DD + d0);
  const float y0 = x.x * norm * g[d0];
  const float y1 = x.y * norm * g[d0 + 1];
  v2f o;
  o.x = y0 * c.x - y1 * s.x;   // (x0,x1) -> x*cos + (-x1,x0)*sin
  o.y = y1 * c.y + y0 * s.y;
  *reinterpret_cast<v2f*>(base + d0) = o;
}

// ---------------------------------------------------------------------------
// Flash attention, fp32 WMMA, online softmax.
// One workgroup = one (b,h) x one 128-row Q block; 8 waves x 16 Q-rows each.
// Q staged once by TDM; K/V streamed in 16-row tiles through double-buffered
// LDS by TDM one tile ahead; P transposed via wave-private LDS slab.
// ---------------------------------------------------------------------------
__global__ __launch_bounds__(256)
void attn_kernel(const float* __restrict__ q, const float* __restrict__ k,
                 const float* __restrict__ v, float* __restrict__ outp) {
  __shared__ __align__(16) float Qs[128 * 68];     // 34 KB, TDM-padded stride
  __shared__ __align__(16) float Ks[2][16 * 68];
  __shared__ __align__(16) float Vs[2][16 * 68];
  __shared__ __align__(16) float Ps[8][16 * 16];   // per-wave C->A transpose

  const int tid  = threadIdx.x;
  const int wave = tid >> 5;
  const int lane = tid & 31;
  const int l16  = lane & 15;
  const int half = lane >> 4;
  const int qb   = blockIdx.x;   // N/128
  const int bh   = blockIdx.y;   // B*H

  const float* qbase = q + (size_t)bh * NN * DD + (size_t)qb * 128 * DD;
  const float* kbase = k + (size_t)bh * NN * DD;
  const float* vbase = v + (size_t)bh * NN * DD;

#if USE_TDM
  if (wave == 0) {     // prime: Q tile + first K/V tiles (in-order TDM queue)
    tdm_load_2d(lds_off(&Qs[0]),    qbase, 64, 128, 64, 5, 3);
    tdm_load_2d(lds_off(&Ks[0][0]), kbase, 64, 16,  64, 5, 3);
    tdm_load_2d(lds_off(&Vs[0][0]), vbase, 64, 16,  64, 5, 3);
  }
#else
#pragma unroll
  for (int i = 0; i < 8; ++i) {
    int idx = tid + i * 256;
    int r = idx >> 4, c4 = (idx & 15) << 2;
    *reinterpret_cast<float4*>(&Qs[r * 68 + c4]) =
        *reinterpret_cast<const float4*>(&qbase[(size_t)r * DD + c4]);
  }
#endif

  v8f acc[4];
#pragma unroll
  for (int t = 0; t < 4; ++t) acc[t] = v8f_zero();
  float mrow[8], lrow[8];
#pragma unroll
  for (int r = 0; r < 8; ++r) { mrow[r] = -__builtin_inff(); lrow[r] = 0.f; }

  for (int kt = 0; kt < NN; kt += 16) {
    const int cur = (kt >> 4) & 1;
#if USE_TDM
    if (wave == 0) {
      if (kt + 16 < NN) {   // prefetch next K/V tile into the other buffer
        tdm_load_2d(lds_off(&Ks[cur ^ 1][0]),
                    &kbase[(size_t)(kt + 16) * DD], 64, 16, 64, 5, 3);
        tdm_load_2d(lds_off(&Vs[cur ^ 1][0]),
                    &vbase[(size_t)(kt + 16) * DD], 64, 16, 64, 5, 3);
        __builtin_amdgcn_s_wait_tensorcnt(2);  // current tiles (+Q) complete
      } else {
        __builtin_amdgcn_s_wait_tensorcnt(0);
      }
    }
#else
    {
      int r = tid >> 4, c4 = (tid & 15) << 2;
      *reinterpret_cast<float4*>(&Ks[cur][r * 68 + c4]) =
          *reinterpret_cast<const float4*>(&kbase[(size_t)(kt + r) * DD + c4]);
      *reinterpret_cast<float4*>(&Vs[cur][r * 68 + c4]) =
          *reinterpret_cast<const float4*>(&vbase[(size_t)(kt + r) * DD + c4]);
    }
#endif
    __syncthreads();   // staged tiles visible to all waves

    // ---- S = Q · K^T over D=64 (16 WMMAs) ----
    v8f s = v8f_zero();
#pragma unroll
    for (int kk = 0; kk < DD; kk += 4) {
      const int kh = kk + 2 * half;
      v2f a = *reinterpret_cast<const v2f*>(&Qs[(wave * 16 + l16) * 68 + kh]);
      v2f b = *reinterpret_cast<const v2f*>(&Ks[cur][l16 * 68 + kh]);
      s = wmma4(a, b, s);
    }

    // ---- online softmax: 16-lane-half row reductions ----
    float afac[8];
#pragma unroll
    for (int r = 0; r < 8; ++r) {
      float val = s[r] * 0.125f;   // 1/sqrt(64)
      float rmax = val;
      rmax = fmaxf(rmax, __shfl_xor(rmax, 1));
      rmax = fmaxf(rmax, __shfl_xor(rmax, 2));
      rmax = fmaxf(rmax, __shfl_xor(rmax, 4));
      rmax = fmaxf(rmax, __shfl_xor(rmax, 8));
      const float nm = fmaxf(mrow[r], rmax);
      const float al = __expf(mrow[r] - nm);
      const float p  = __expf(val - nm);
      float rs = p;
      rs += __shfl_xor(rs, 1);  rs += __shfl_xor(rs, 2);
      rs += __shfl_xor(rs, 4);  rs += __shfl_xor(rs, 8);
      mrow[r] = nm;
      lrow[r] = lrow[r] * al + rs;
      afac[r] = al;
      s[r] = p;
    }
#pragma unroll
    for (int t = 0; t < 4; ++t)
#pragma unroll
      for (int r = 0; r < 8; ++r) acc[t][r] = acc[t][r] * afac[r];

    // ---- P: C-layout -> A-layout via wave-private LDS ----
#pragma unroll
    for (int r = 0; r < 8; ++r)
      Ps[wave][(r + half * 8) * 16 + l16] = s[r];
    // same-wave LDS ops are in order; backend inserts s_wait_dscnt

    // ---- O += P · V  (16 WMMAs) ----
#pragma unroll
    for (int j = 0; j < 4; ++j) {
      const int kr = 4 * j + 2 * half;
      v2f a = *reinterpret_cast<const v2f*>(&Ps[wave][l16 * 16 + kr]);
#pragma unroll
      for (int t = 0; t < 4; ++t) {
        v2f b;
        b.x = Vs[cur][kr * 68 + t * 16 + l16];
        b.y = Vs[cur][(kr + 1) * 68 + t * 16 + l16];
        acc[t] = wmma4(a, b, acc[t]);
      }
    }
    __syncthreads();   // reads done before buffer `cur` is refilled
  }

  // ---- normalize and store to (B,N,H*D) ----
  const int b  = bh >> 4;
  const int hh = bh & 15;
#pragma unroll
  for (int r = 0; r < 8; ++r) {
    const float inv = 1.0f / lrow[r];
    const int n = qb * 128 + wave * 16 + r + half * 8;
#pragma unroll
    for (int t = 0; t < 4; ++t) {
      const int d = t * 16 + l16;
      outp[(((size_t)b * NN + n) * CC) + hh * DD + d] = acc[t][r] * inv;
    }
  }
}

// ---------------------------------------------------------------------------
extern "C" void kernel_launch(void* const* d_in, const int* in_sizes, int n_in,
                              void* d_out, int out_size, void* d_ws,
                              size_t ws_size, hipStream_t stream) {
  (void)in_sizes; (void)n_in; (void)out_size; (void)ws_size;
  const float* x      = (const float*)d_in[0];
  const float* cosb   = (const float*)d_in[1];
  const float* sinb   = (const float*)d_in[2];
  const float* w_qkv  = (const float*)d_in[3];
  const float* w_proj = (const float*)d_in[4];
  const float* qg     = (const float*)d_in[5];
  const float* kg     = (const float*)d_in[6];
  float* out = (float*)d_out;

  float* ws = (float*)d_ws;
  float* qw = ws;                       // (B,H,N,D)
  float* kw = ws + QKV_ELEMS;
  float* vw = ws + 2 * QKV_ELEMS;
  float* aw = ws + 3 * QKV_ELEMS;       // (B,N,C) attention output

  // 1) QKV projection, scattered to (B,H,N,D) per tensor
  gemm_kernel<0><<<dim3(BN / 64, FQKV / 128), 256, 0, stream>>>(
      x, w_qkv, qw, kw, vw);
  // 2) RMSNorm + RoPE on q and k in place
  rmsrope_kernel<<<(2 * BHN) / 8, 256, 0, stream>>>(qw, kw, cosb, sinb, qg, kg);
  // 3) Flash attention
  attn_kernel<<<dim3(NN / 128, BB * HH), 256, 0, stream>>>(qw, kw, vw, aw);
  // 4) Output projection
  gemm_kernel<1><<<dim3(BN / 64, CC / 128), 256, 0, stream>>>(
      aw, w_proj, out, nullptr, nullptr);
}